// ChurnLoss_14491219657064
// MI455X (gfx1250) — compile-verified
//
#include <hip/hip_runtime.h>
#include <hip/hip_bf16.h>

typedef __attribute__((ext_vector_type(2))) float v2f;
typedef __attribute__((ext_vector_type(8))) float v8f;

#define BLOCK 256
#define EPS   1e-5f
#define LN2f    0.69314718055994530942f   // ln(2)
#define LOG2Ef  1.44269504088896340736f   // log2(e)

// Full-precision wave32 sum using V_WMMA_F32_16X16X4_F32 with an all-ones B
// matrix (B layout irrelevant since every element is 1.0).
// Stage 1: A[m,0]=acc[lane m] (lanes 0-15), A[m,2]=acc[lane m+16] (lanes 16-31),
//          other K slots 0  ->  D[m,n] = acc[m] + acc[m+16]  (row-sums R[m]).
// D layout: VGPR r = R[r] (lanes 0-15) / R[r+8] (lanes 16-31), constant over n.
// Stage 2: feed D vgpr pairs back as A, accumulate over 4 WMMAs ->
//          every element of the final D equals the 32-lane total.
__device__ __forceinline__ float wave_sum_wmma(float x) {
  v2f a;   a[0] = x;    a[1] = 0.0f;
  v2f one; one[0] = 1.0f; one[1] = 1.0f;
  v8f z = {};
  v8f d1 = __builtin_amdgcn_wmma_f32_16x16x4_f32(false, a, false, one, (short)0, z, false, false);
  v8f acc = {};
  v2f a2;
  a2[0] = d1[0]; a2[1] = d1[1];
  acc = __builtin_amdgcn_wmma_f32_16x16x4_f32(false, a2, false, one, (short)0, acc, false, false);
  a2[0] = d1[2]; a2[1] = d1[3];
  acc = __builtin_amdgcn_wmma_f32_16x16x4_f32(false, a2, false, one, (short)0, acc, false, false);
  a2[0] = d1[4]; a2[1] = d1[5];
  acc = __builtin_amdgcn_wmma_f32_16x16x4_f32(false, a2, false, one, (short)0, acc, false, false);
  a2[0] = d1[6]; a2[1] = d1[7];
  acc = __builtin_amdgcn_wmma_f32_16x16x4_f32(false, a2, false, one, (short)0, acc, false, false);
  return acc[0];
}

// Deterministic 256-thread block sum (valid in thread 0).
__device__ __forceinline__ float block_sum(float x) {
  __shared__ float ws[BLOCK / 32];
  const float w = wave_sum_wmma(x);      // EXEC all-ones here: no divergence
  const int lane = threadIdx.x & 31;
  const int wid  = threadIdx.x >> 5;
  if (lane == 0) ws[wid] = w;
  __syncthreads();
  float s = 0.0f;
  if (threadIdx.x == 0) {
#pragma unroll
    for (int i = 0; i < BLOCK / 32; ++i) s += ws[i];
  }
  return s;
}

// One block per sequence; lane handles event s+tid (+stride for long seqs).
// Interior term, fused to one v_log_f32 + two v_rcp_f32 (raw HW trans ops;
// all arguments are strictly normal-range so no edge-case code needed):
//   -log(1-p+eps) + log(tau) + (dt_next+eps)/tau
//   = ln2 * log2(tau * rcp(1-p+eps)) + (dt_next+eps) * rcp(tau)
__global__ __launch_bounds__(BLOCK)
void churn_seq_kernel(const float* __restrict__ dt,
                      const float* __restrict__ tau,
                      const float* __restrict__ p,
                      const float* __restrict__ t_to_now,
                      const int*   __restrict__ offsets,
                      float* __restrict__ partials) {
  const int b = blockIdx.x;
  const int s = offsets[b];
  const int e = offsets[b + 1];
  // terminal exponent scale, uniform per block: -(t_to_now+eps)*log2(e)
  const float tn2 = -(t_to_now[b] + EPS) * LOG2Ef;

  float acc = 0.0f;
  for (int j = s + (int)threadIdx.x; j < e; j += BLOCK) {
    const float tj = tau[j];
    const float rt = __builtin_amdgcn_rcpf(tj);              // v_rcp_f32
    if (j == e - 1) {
      // terminal: -log((1-p)*exp(-(t+eps)/tau) + p + eps)
      const float pl   = p[j];
      const float surv = __builtin_amdgcn_exp2f(tn2 * rt);   // v_exp_f32
      const float arg  = fmaf(1.0f - pl, surv, pl + EPS);
      acc -= LN2f * __builtin_amdgcn_logf(arg);              // v_log_f32
    } else if (j > s) {
      const float pj = p[j];
      const float dn = dt[j + 1] + EPS;                      // shifted, coalesced
      const float r1 = __builtin_amdgcn_rcpf(1.0f - pj + EPS);
      acc = fmaf(LN2f, __builtin_amdgcn_logf(tj * r1), fmaf(dn, rt, acc));
    }
  }

  const float bs = block_sum(acc);
  if (threadIdx.x == 0) partials[b] = bs;
}

// Single-block deterministic final reduction of the per-sequence partials.
__global__ __launch_bounds__(BLOCK)
void churn_final_kernel(const float* __restrict__ partials, int nb,
                        float inv_n, float* __restrict__ out) {
  float acc = 0.0f;
  for (int i = (int)threadIdx.x; i < nb; i += BLOCK) acc += partials[i];
  const float bs = block_sum(acc);
  if (threadIdx.x == 0) out[0] = bs * inv_n;
}

extern "C" void kernel_launch(void* const* d_in, const int* in_sizes, int n_in,
                              void* d_out, int out_size, void* d_ws, size_t ws_size,
                              hipStream_t stream) {
  const float* dt       = (const float*)d_in[0];
  const float* tau      = (const float*)d_in[1];
  const float* p        = (const float*)d_in[2];
  const float* t_to_now = (const float*)d_in[3];
  const int*   offsets  = (const int*)d_in[4];

  const int n = in_sizes[0];   // total events
  const int B = in_sizes[3];   // number of sequences (t_to_now length)

  float* partials = (float*)d_ws;  // B floats of scratch (8192 * 4 = 32 KB)

  churn_seq_kernel<<<B, BLOCK, 0, stream>>>(dt, tau, p, t_to_now, offsets, partials);
  churn_final_kernel<<<1, BLOCK, 0, stream>>>(partials, B, 1.0f / (float)n,
                                              (float*)d_out);
  (void)n_in; (void)out_size; (void)ws_size;
}